// IPGLayer_67164698575278
// MI455X (gfx1250) — compile-verified
//
#include <hip/hip_runtime.h>
#include <hip/hip_bf16.h>
#include <math.h>

// ---------------- CDNA5 WMMA types ----------------
typedef __attribute__((ext_vector_type(2))) float v2f;
typedef __attribute__((ext_vector_type(8))) float v8f;

// Problem constants (reference: B=2, C=64, H=W=64, win=17, MAXC=16)
#define CCH   64
#define HH    64
#define WW    64
#define PIX   (HH*WW)          // 4096
#define WIN   17
#define HALF  8
#define NNB   (WIN*WIN)        // 289
#define DHID  128              // 2*C

// ---------- kernel 1: 2x average downsample: xd(B,C,32,32) ----------
__global__ void k_down(const float* __restrict__ x, float* __restrict__ xd, int total) {
    int t = blockIdx.x * blockDim.x + threadIdx.x;
    if (t >= total) return;
    int j = t & 31;
    int i = (t >> 5) & 31;
    int bc = t >> 10;                 // b*64 + c
    const float* p = x + ((size_t)bc * HH + 2 * i) * WW + 2 * j;
    xd[t] = 0.25f * (p[0] + p[1] + p[WW] + p[WW + 1]);
}

// bilinear upsample index helper (matches jnp: src=(o+0.5)*in/out-0.5)
__device__ inline void upidx(int o, int& i0, int& i1, float& t) {
    float src = (o + 0.5f) * 0.5f - 0.5f;
    float f = floorf(src);
    t = src - f;
    int i = (int)f;
    i0 = min(max(i, 0), 31);
    i1 = min(max(i + 1, 0), 31);
}

// ---------- kernel 2: df(B,H,W) = sum_c |x - upsample(xd)| ----------
__global__ void k_diff(const float* __restrict__ x, const float* __restrict__ xd,
                       float* __restrict__ df, int total) {
    int t = blockIdx.x * blockDim.x + threadIdx.x;
    if (t >= total) return;
    int w = t & (WW - 1);
    int h = (t >> 6) & (HH - 1);
    int b = t >> 12;
    int i0, i1, j0, j1; float th, tw;
    upidx(h, i0, i1, th);
    upidx(w, j0, j1, tw);
    float acc = 0.f;
    for (int c = 0; c < CCH; ++c) {
        const float* d = xd + ((size_t)(b * CCH + c)) * (32 * 32);
        float v00 = d[i0 * 32 + j0], v01 = d[i0 * 32 + j1];
        float v10 = d[i1 * 32 + j0], v11 = d[i1 * 32 + j1];
        float xu = (v00 * (1.f - tw) + v01 * tw) * (1.f - th) +
                   (v10 * (1.f - tw) + v11 * tw) * th;
        float xv = x[(((size_t)(b * CCH + c)) << 12) + (h << 6) + w];
        acc += fabsf(xv - xu);
    }
    df[t] = acc;
}

// ---------- kernel 3: per-batch min/max of df ----------
__global__ void k_minmax(const float* __restrict__ df, float* __restrict__ mm) {
    __shared__ float smin[256], smax[256];
    int b = blockIdx.x, t = threadIdx.x;
    float mn = 3.4e38f, mx = -3.4e38f;
    for (int i = t; i < PIX; i += 256) {
        float v = df[b * PIX + i];
        mn = fminf(mn, v); mx = fmaxf(mx, v);
    }
    smin[t] = mn; smax[t] = mx;
    __syncthreads();
    for (int s = 128; s; s >>= 1) {
        if (t < s) { smin[t] = fminf(smin[t], smin[t + s]); smax[t] = fmaxf(smax[t], smax[t + s]); }
        __syncthreads();
    }
    if (t == 0) { mm[b * 2] = smin[0]; mm[b * 2 + 1] = smax[0]; }
}

// ---------- kernel 4: connection counts k in [1,16] ----------
__global__ void k_counts(const float* __restrict__ df, const float* __restrict__ mm,
                         int* __restrict__ kc, int total) {
    int t = blockIdx.x * blockDim.x + threadIdx.x;
    if (t >= total) return;
    int b = t >> 12;
    float dmin = mm[b * 2], dmax = mm[b * 2 + 1];
    float dn = (df[t] - dmin) / (dmax - dmin + 1e-8f);
    kc[t] = 1 + (int)rintf(dn * 15.0f);
}

// ---------- kernel 5: x_flat / xn (L2-normalized) / xln (LayerNorm) pixel-major ----------
__global__ void k_prep(const float* __restrict__ x, const float* __restrict__ gam,
                       const float* __restrict__ bet, float* __restrict__ xflat,
                       float* __restrict__ xn, float* __restrict__ xln, int total) {
    int t = blockIdx.x * blockDim.x + threadIdx.x;
    if (t >= total) return;
    int b = t >> 12;
    int p = t & (PIX - 1);
    float sum = 0.f, sq = 0.f;
    float* fo = xflat + (size_t)t * CCH;
    for (int c = 0; c < CCH; ++c) {
        float v = x[(((size_t)(b * CCH + c)) << 12) + p];
        fo[c] = v;
        sum += v; sq += v * v;
    }
    float mu = sum * (1.f / CCH);
    float var = sq * (1.f / CCH) - mu * mu;
    float rs = rsqrtf(var + 1e-5f);
    float inrm = 1.f / fmaxf(sqrtf(sq), 1e-12f);
    float* no = xn + (size_t)t * CCH;
    float* lo = xln + (size_t)t * CCH;
    for (int c = 0; c < CCH; ++c) {
        float v = fo[c];
        no[c] = v * inrm;
        lo[c] = (v - mu) * rs * gam[c] + bet[c];
    }
}

// ---------- kernel 6: similarity + top-16 + softmax aggregate + residual LN ----------
// one wave32 per pixel, 8 waves (256 threads) per block
__global__ __launch_bounds__(256) void k_simagg(const float* __restrict__ xn,
                                                const float* __restrict__ xflat,
                                                const float* __restrict__ xln,
                                                const int* __restrict__ kcnt,
                                                float* __restrict__ enh) {
    __shared__ __align__(16) float s_xn[8][CCH];
    __shared__ float s_sim[8][NNB + 7];
    __shared__ float s_tv[8][16];
    __shared__ int   s_tq[8][16];
    int lane = threadIdx.x & 31;
    int wv = threadIdx.x >> 5;
    int g = blockIdx.x * 8 + wv;          // global pixel id over B*PIX
    int b = g >> 12;
    int p = g & (PIX - 1);
    int py = p >> 6, px = p & 63;

    const float* me = xn + (size_t)g * CCH;
    s_xn[wv][lane] = me[lane];
    s_xn[wv][lane + 32] = me[lane + 32];

    // similarities, lanes stride over 289 window offsets
    for (int j = lane; j < NNB; j += 32) {
        int dy = j / WIN - HALF, dx = j % WIN - HALF;
        int ny = py + dy, nx = px + dx;
        bool valid = (ny >= 0) & (ny < HH) & (nx >= 0) & (nx < WW);
        int cy = min(max(ny, 0), HH - 1), cx = min(max(nx, 0), WW - 1);
        int q = cy * WW + cx;
        const float4* nb = (const float4*)(xn + ((size_t)(b << 12) + q) * CCH);
        __builtin_prefetch(xflat + ((size_t)(b << 12) + q) * CCH, 0, 3);
        const float4* mine = (const float4*)(&s_xn[wv][0]);
        float dot = 0.f;
        #pragma unroll
        for (int c4 = 0; c4 < CCH / 4; ++c4) {
            float4 a = mine[c4];
            float4 bb = nb[c4];
            dot += a.x * bb.x + a.y * bb.y + a.z * bb.z + a.w * bb.w;
        }
        s_sim[wv][j] = valid ? dot : -1e30f;
    }

    // iterative cooperative argmax: top-16 (sorted descending, first-index ties)
    for (int i = 0; i < 16; ++i) {
        float bv = -3.4e38f; int bj = 0;
        for (int j = lane; j < NNB; j += 32) {
            float v = s_sim[wv][j];
            if (v > bv) { bv = v; bj = j; }
        }
        for (int off = 16; off; off >>= 1) {
            float ov = __shfl_xor(bv, off, 32);
            int   oj = __shfl_xor(bj, off, 32);
            if (ov > bv || (ov == bv && oj < bj)) { bv = ov; bj = oj; }
        }
        if (lane == 0) {
            s_sim[wv][bj] = -3.4e38f;
            int dy = bj / WIN - HALF, dx = bj % WIN - HALF;
            int cy = min(max(py + dy, 0), HH - 1), cx = min(max(px + dx, 0), WW - 1);
            s_tq[wv][i] = cy * WW + cx;
            s_tv[wv][i] = bv;
        }
    }

    // masked softmax weights over first-k entries
    int kc = kcnt[g];
    float w = 0.f;
    if (lane < 16) w = (lane < kc) ? __expf(s_tv[wv][lane]) : 0.f;
    float ws = w;
    for (int off = 8; off; off >>= 1) ws += __shfl_xor(ws, off, 32);
    float wtot = __shfl(ws, 0, 32);      // sum of lanes 0..15 broadcast to all
    if (lane < 16) s_tv[wv][lane] = w / wtot;

    // weighted aggregate of raw features, add LayerNorm residual
    float a0 = 0.f, a1 = 0.f;
    for (int i = 0; i < 16; ++i) {
        float wi = s_tv[wv][i];
        int q = s_tq[wv][i];
        const float* f = xflat + ((size_t)(b << 12) + q) * CCH;
        a0 += wi * f[lane];
        a1 += wi * f[lane + 32];
    }
    size_t o = (size_t)g * CCH;
    enh[o + lane]      = a0 + xln[o + lane];
    enh[o + lane + 32] = a1 + xln[o + lane + 32];
}

// ---------- kernel 7: h1 = relu(enh @ w1^T + b1), WMMA f32 16x16x4 ----------
// enh: (M,64) row-major ; w1: (128,64) row-major ; h1: (M,128)
__global__ __launch_bounds__(256) void k_ffn1(const float* __restrict__ enh,
                                              const float* __restrict__ w1,
                                              const float* __restrict__ b1,
                                              float* __restrict__ h1) {
    int wid = (blockIdx.x * blockDim.x + threadIdx.x) >> 5;
    int lane = threadIdx.x & 31;
    int mt = wid >> 3;            // 8 N-tiles across DHID=128
    int nt = wid & 7;
    int row = mt * 16 + (lane & 15);
    int col = nt * 16 + (lane & 15);
    int ksel = (lane >> 4) << 1;  // lanes 0-15 -> K0,K1 ; lanes 16-31 -> K2,K3
    v8f acc = {};
    #pragma unroll
    for (int k0 = 0; k0 < CCH; k0 += 4) {
        v2f a = *(const v2f*)(enh + (size_t)row * CCH + k0 + ksel);
        v2f bf = *(const v2f*)(w1 + (size_t)col * CCH + k0 + ksel);   // B[k][n] = w1[n][k]
        acc = __builtin_amdgcn_wmma_f32_16x16x4_f32(false, a, false, bf,
                                                    (short)0, acc, false, false);
    }
    float bias = b1[col];
    int mbase = mt * 16 + ((lane >> 4) << 3);
    #pragma unroll
    for (int r = 0; r < 8; ++r) {
        float v = acc[r] + bias;
        h1[(size_t)(mbase + r) * DHID + col] = v > 0.f ? v : 0.f;   // lanes vary col: coalesced
    }
}

// ---------- kernel 8: out = enh + (h1 @ w2^T + b2), LDS-transposed coalesced BCHW store ----
// h1: (M,128) ; w2: (64,128) row-major ; out: (B,64,64,64)
__global__ __launch_bounds__(256) void k_ffn2(const float* __restrict__ h1,
                                              const float* __restrict__ enh,
                                              const float* __restrict__ w2,
                                              const float* __restrict__ b2,
                                              float* __restrict__ out) {
    __shared__ float s_t[8][16][17];   // per-wave 16x16 tile (+1 pad vs bank conflicts)
    int wid = (blockIdx.x * blockDim.x + threadIdx.x) >> 5;
    int lane = threadIdx.x & 31;
    int wv = (threadIdx.x >> 5) & 7;
    int mt = wid >> 2;            // 4 N-tiles across C=64
    int nt = wid & 3;
    int row = mt * 16 + (lane & 15);
    int col = nt * 16 + (lane & 15);
    int ksel = (lane >> 4) << 1;
    v8f acc = {};
    #pragma unroll
    for (int k0 = 0; k0 < DHID; k0 += 4) {
        v2f a = *(const v2f*)(h1 + (size_t)row * DHID + k0 + ksel);
        v2f bf = *(const v2f*)(w2 + (size_t)col * DHID + k0 + ksel);  // B[k][n] = w2[n][k]
        acc = __builtin_amdgcn_wmma_f32_16x16x4_f32(false, a, false, bf,
                                                    (short)0, acc, false, false);
    }
    // epilogue part 1: bias + enh residual while lanes are column-contiguous, park in LDS
    float bias = b2[col];
    int mhalf = (lane >> 4) << 3;     // 0 or 8
    #pragma unroll
    for (int r = 0; r < 8; ++r) {
        int m = mhalf + r;                               // tile-local row
        int rm = mt * 16 + m;                            // global row over B*PIX
        float v = enh[(size_t)rm * CCH + col] + acc[r] + bias;   // coalesced read
        s_t[wv][m][lane & 15] = v;
    }
    // epilogue part 2: transposed store — lanes vary over pixel index p (coalesced BCHW).
    // Tile rows never straddle the batch boundary (4096 % 16 == 0).
    int rm0 = mt * 16;
    int b = rm0 >> 12;
    int p0 = rm0 & (PIX - 1);
    int m = lane & 15;                 // pixel within tile
    #pragma unroll
    for (int r = 0; r < 8; ++r) {
        int n = (lane >> 4) * 8 + r;   // channel within tile; half-waves cover n=0..7 / 8..15
        int ch = nt * 16 + n;
        out[((size_t)(b * CCH + ch) << 12) + p0 + m] = s_t[wv][m][n];
    }
}

extern "C" void kernel_launch(void* const* d_in, const int* in_sizes, int n_in,
                              void* d_out, int out_size, void* d_ws, size_t ws_size,
                              hipStream_t stream) {
    const float* x   = (const float*)d_in[0];
    const float* gam = (const float*)d_in[1];
    const float* bet = (const float*)d_in[2];
    const float* w1  = (const float*)d_in[3];
    const float* b1  = (const float*)d_in[4];
    const float* w2  = (const float*)d_in[5];
    const float* b2  = (const float*)d_in[6];
    float* out = (float*)d_out;

    const int B = in_sizes[0] / (CCH * PIX);
    const int M = B * PIX;

    char* ws = (char*)d_ws;
    auto alloc = [&](size_t bytes) -> char* {
        char* p = ws;
        ws += (bytes + 255) & ~(size_t)255;
        return p;
    };
    float* xd    = (float*)alloc((size_t)B * CCH * 32 * 32 * sizeof(float));
    float* df    = (float*)alloc((size_t)B * PIX * sizeof(float));
    float* mm    = (float*)alloc((size_t)B * 2 * sizeof(float));
    int*   kc    = (int*)  alloc((size_t)B * PIX * sizeof(int));
    float* xflat = (float*)alloc((size_t)M * CCH * sizeof(float));
    float* xn    = (float*)alloc((size_t)M * CCH * sizeof(float));
    float* xln   = (float*)alloc((size_t)M * CCH * sizeof(float));
    float* enh   = (float*)alloc((size_t)M * CCH * sizeof(float));
    float* h1    = (float*)alloc((size_t)M * DHID * sizeof(float));

    int tDown = B * CCH * 32 * 32;
    k_down<<<(tDown + 255) / 256, 256, 0, stream>>>(x, xd, tDown);

    int tPix = B * PIX;
    k_diff<<<(tPix + 255) / 256, 256, 0, stream>>>(x, xd, df, tPix);
    k_minmax<<<B, 256, 0, stream>>>(df, mm);
    k_counts<<<(tPix + 255) / 256, 256, 0, stream>>>(df, mm, kc, tPix);
    k_prep<<<(tPix + 255) / 256, 256, 0, stream>>>(x, gam, bet, xflat, xn, xln, tPix);

    k_simagg<<<tPix / 8, 256, 0, stream>>>(xn, xflat, xln, kc, enh);

    // GEMM1: (M/16) M-tiles x 8 N-tiles, 8 waves/block
    k_ffn1<<<(M / 16) * 8 / 8, 256, 0, stream>>>(enh, w1, b1, h1);
    // GEMM2: (M/16) M-tiles x 4 N-tiles, 8 waves/block
    k_ffn2<<<(M / 16) * 4 / 8, 256, 0, stream>>>(h1, enh, w2, b2, out);
}